// _HalfLatticeHopAdapter_78469052498784
// MI455X (gfx1250) — compile-verified
//
#include <hip/hip_runtime.h>

// ---------------------------------------------------------------------------
// Half-lattice hop operator (Dirac-like stencil), MI455X / gfx1250.
//
// Roofline: K-matrix stream = 302 MB (read once, > 192 MB L2) dominates;
// AI ~1 flop/byte => HBM-bandwidth-bound (~13 us floor @ 23.3 TB/s).
//   * K      : non-temporal global_load_b128 (keep L2 for the 3 MB psi field)
//   * psi    : (t,z)-plane + 4 halo planes staged to LDS by the Tensor Data
//              Mover (tensor_load_to_lds, TENSORcnt) -- each plane-part is a
//              contiguous 6144 B tile; DMA frees all wave issue slots for K
//   * compute: thread-per-site fp32 FMA chains, 128-thr (4-wave) blocks,
//              rolled direction loop (no spills, ~1/4 code size)
//
// Round-4 fix: this toolchain's __builtin_amdgcn_tensor_load_to_lds is the
// 6-arg clang-23 form (uint32x4, int32x8, int32x4, int32x4, int32x8, i32).
// ---------------------------------------------------------------------------

typedef float v4f __attribute__((ext_vector_type(4)));
typedef unsigned v4u __attribute__((ext_vector_type(4)));
typedef int v4i __attribute__((ext_vector_type(4)));
typedef int v8i __attribute__((ext_vector_type(8)));

#define Td   16
#define Zd   16
#define Yd   16
#define XHd  8
#define Dd   12
#define NSITE (Td * Zd * Yd * XHd)            /* 32768 */
#define MUSTR ((size_t)NSITE * Dd * Dd)       /* per-mu element stride in K */
#define DIAGC 4.5f

#define PLANE_BYTES (128 * Dd * 4)            /* 6144 B: one plane, re or im */
#define LDS_IM_BASE (5 * PLANE_BYTES)         /* im region starts at 30720 B */

// One 1-D TDM descriptor: copy 6144 contiguous bytes global -> LDS.
// D# per CDNA5 ISA ch.8: group0 {count=1, lds_addr, global_addr, type=2},
// group1 {data_size=4B, tensor_dim0=tile_dim0=1536 dwords, dim1=1,
// dim0_stride=1536, no padding / no cluster multicast}, groups 2/3 zero.
__device__ __forceinline__ void tdm_copy_plane(unsigned lds_byte, const void* gptr) {
  const unsigned long long ga = (unsigned long long)gptr;
  v4u g0;
  g0[0] = 1u;                                            // count=1, user mode
  g0[1] = lds_byte;                                      // lds_addr
  g0[2] = (unsigned)ga;                                  // global_addr[31:0]
  g0[3] = (unsigned)((ga >> 32) & 0x01FFFFFFu)           // global_addr[56:32]
          | (2u << 30);                                  // type=2 ("image")
  v8i g1;
  g1[0] = 0x20000;                                       // data_size=2 (4 B)
  g1[1] = (int)(1536u << 16);                            // tensor_dim0 = 1536
  g1[2] = (int)(1u << 16);                               // tensor_dim1 = 1
  g1[3] = (int)(1536u << 16);                            // tile_dim0 = 1536
  g1[4] = 0;                                             // tile_dim1/2 unused
  g1[5] = 1536;                                          // tensor_dim0_stride
  g1[6] = 0;
  g1[7] = 0;
  v4i gz4 = {0, 0, 0, 0};
  v8i gz8 = {0, 0, 0, 0, 0, 0, 0, 0};
  // clang-23 / therock-10.0 6-arg form (probe-confirmed arity on this lane)
  __builtin_amdgcn_tensor_load_to_lds(g0, g1, gz4, gz4, gz8, 0);
}

// acc += K * v (complex 12x12 mat-vec). K streamed non-temporally as b128,
// psi vector read from LDS as 3+3 b128 chunks (48 B slots, 16 B aligned).
// Fully unrolled inside so accr/acci are only indexed by constants.
__device__ __forceinline__ void matacc(const float* kr,
                                       const float* ki,
                                       const float* vre,  // 12 LDS floats
                                       const float* vim,  // 12 LDS floats
                                       float* accr, float* acci) {
  const v4f* vr = (const v4f*)vre;
  const v4f* vi = (const v4f*)vim;
  v4f vr4[3], vi4[3];
#pragma unroll
  for (int c = 0; c < 3; ++c) { vr4[c] = vr[c]; vi4[c] = vi[c]; }
#pragma unroll
  for (int i = 0; i < Dd; ++i) {
    float ar = 0.f, ai = 0.f;
#pragma unroll
    for (int c = 0; c < 3; ++c) {
      v4f r4 = __builtin_nontemporal_load((const v4f*)(kr + i * Dd + 4 * c));
      v4f s4 = __builtin_nontemporal_load((const v4f*)(ki + i * Dd + 4 * c));
#pragma unroll
      for (int q = 0; q < 4; ++q) {
        ar += r4[q] * vr4[c][q] - s4[q] * vi4[c][q];
        ai += r4[q] * vi4[c][q] + s4[q] * vr4[c][q];
      }
    }
    accr[i] += ar;
    acci[i] += ai;
  }
}

__global__ __launch_bounds__(128)
void dslash_kernel(const float* __restrict__ psi_re, const float* __restrict__ psi_im,
                   const float* __restrict__ Kfr, const float* __restrict__ Kfi,
                   const float* __restrict__ Kbr, const float* __restrict__ Kbi,
                   float* __restrict__ out_re, float* __restrict__ out_im) {
  // psi re: 5 planes x 128 sites x 12 floats, then im: same. 60 KB total.
  __shared__ float lds_re[5 * 128 * Dd];
  __shared__ float lds_im[5 * 128 * Dd];

  // Escape both LDS arrays so the TDM writes (not modeled by alias analysis)
  // cannot lead to the reads being folded to undef and the mat-vec DCE'd.
  {
    float* p = lds_re;
    float* q = lds_im;
    asm volatile("" : "+v"(p), "+v"(q)::"memory");
  }

  const int plane = blockIdx.x;          // t*16 + z
  const int t = plane >> 4, z = plane & 15;
  const int s = threadIdx.x;             // y*8 + x
  const int y = s >> 3, x = s & 7;

  const int tp = (t + 1) & 15, tm = (t + 15) & 15;
  const int zp = (z + 1) & 15, zm = (z + 15) & 15;

  // plane ids (t'*16+z') for: self, t+1, t-1, z+1, z-1
  int P[5];
  P[0] = t  * 16 + z;
  P[1] = tp * 16 + z;
  P[2] = tm * 16 + z;
  P[3] = t  * 16 + zp;
  P[4] = t  * 16 + zm;

  // --- TDM staging: wave 0 issues 10 descriptors (5 planes x re/im). ---
  // All operands are workgroup-uniform; TDM ignores EXEC; TENSORcnt tracks.
  if ((threadIdx.x >> 5) == 0) {
#pragma unroll
    for (int pl = 0; pl < 5; ++pl) {
      const unsigned pb = (unsigned)(pl * PLANE_BYTES);
      tdm_copy_plane(pb,               (const char*)psi_re + (size_t)P[pl] * PLANE_BYTES);
      tdm_copy_plane(LDS_IM_BASE + pb, (const char*)psi_im + (size_t)P[pl] * PLANE_BYTES);
    }
    __builtin_amdgcn_s_wait_tensorcnt(0);
  }
  __syncthreads();

  const int g0 = (P[0] * 16 + y) * 8 + x;   // this thread's global site

  // per-mu LDS source slots (site index within the 5-plane staging buffer)
  const int r  = (t + z + y) & 1;
  const int xf = r ? ((x + 1) & 7) : x;   // staggered fwd hop only on r==1
  const int xb = r ? x : ((x + 7) & 7);   // staggered bwd hop only on r==0
  int fslot[4], bslot[4];
  fslot[0] = 1 * 128 + s;                    bslot[0] = 2 * 128 + s;
  fslot[1] = 3 * 128 + s;                    bslot[1] = 4 * 128 + s;
  fslot[2] = ((y + 1)  & 15) * 8 + x;        bslot[2] = ((y + 15) & 15) * 8 + x;
  fslot[3] = y * 8 + xf;                     bslot[3] = y * 8 + xb;

  const size_t kb = (size_t)g0 * (Dd * Dd);
  const float* pfr = Kfr + kb;
  const float* pfi = Kfi + kb;
  const float* pbr = Kbr + kb;
  const float* pbi = Kbi + kb;

  float accr[Dd], acci[Dd];
#pragma unroll
  for (int i = 0; i < Dd; ++i) { accr[i] = 0.f; acci[i] = 0.f; }

  // Rolled direction loop: keeps live state bounded (no spills) and code
  // size ~1/4 of the full unroll. Body = 2 fully-unrolled complex mat-vecs.
#pragma clang loop unroll(disable)
  for (int mu = 0; mu < 4; ++mu) {
    matacc(pfr, pfi, &lds_re[fslot[mu] * Dd], &lds_im[fslot[mu] * Dd], accr, acci);
    matacc(pbr, pbi, &lds_re[bslot[mu] * Dd], &lds_im[bslot[mu] * Dd], accr, acci);
    pfr += MUSTR; pfi += MUSTR; pbr += MUSTR; pbi += MUSTR;
  }

  // out = DIAG * psi - 0.5 * acc ; non-temporal b128 stores (write-once stream)
  const float* vsr = &lds_re[s * Dd];
  const float* vsi = &lds_im[s * Dd];
  float* orp = out_re + (size_t)g0 * Dd;
  float* oip = out_im + (size_t)g0 * Dd;
#pragma unroll
  for (int c = 0; c < 3; ++c) {
    v4f vo_r, vo_i;
#pragma unroll
    for (int q = 0; q < 4; ++q) {
      const int i = 4 * c + q;
      vo_r[q] = DIAGC * vsr[i] - 0.5f * accr[i];
      vo_i[q] = DIAGC * vsi[i] - 0.5f * acci[i];
    }
    __builtin_nontemporal_store(vo_r, (v4f*)(orp + 4 * c));
    __builtin_nontemporal_store(vo_i, (v4f*)(oip + 4 * c));
  }
}

extern "C" void kernel_launch(void* const* d_in, const int* in_sizes, int n_in,
                              void* d_out, int out_size, void* d_ws, size_t ws_size,
                              hipStream_t stream) {
  const float* psi_re = (const float*)d_in[0];
  const float* psi_im = (const float*)d_in[1];
  const float* Kfr    = (const float*)d_in[2];
  const float* Kfi    = (const float*)d_in[3];
  const float* Kbr    = (const float*)d_in[4];
  const float* Kbi    = (const float*)d_in[5];

  float* out_re = (float*)d_out;                       // tuple order: re ...
  float* out_im = out_re + (size_t)NSITE * Dd;         // ... then im

  dslash_kernel<<<dim3(Td * Zd), dim3(Yd * XHd), 0, stream>>>(
      psi_re, psi_im, Kfr, Kfi, Kbr, Kbi, out_re, out_im);
}